// Revisit_RDLoss_61572651155970
// MI455X (gfx1250) — compile-verified
//
#include <hip/hip_runtime.h>
#include <math.h>

typedef __attribute__((ext_vector_type(2))) float v2f;
typedef __attribute__((ext_vector_type(8))) float v8f;

// ---- workspace layout (in floats) ----
#define WS_PM    0        // 3*16*64 = 3072 floats (partial row max)
#define WS_PS    3072     // 3072 floats (partial row sumexp)
#define WS_ROWC  6144     // 3*16 = 48 floats (m + log s per row)
#define WS_G     6208     // 3*256 = 768 floats (Gram matrices)
#define WS_REC   6976     // 3 floats
#define WS_CON   6980     // 3*16*3 = 144 floats (dot, |x|^2, |y|^2)
#define WS_END   7124

// ---------------- zero the accumulated regions ----------------
__global__ void init_ws(float* ws) {
  for (int i = threadIdx.x; i < (WS_END - WS_G); i += blockDim.x)
    ws[WS_G + i] = 0.0f;
}

// ---------------- softmax stats stage 1: per-chunk online (m, s) ----------------
__global__ void sm1(const float* __restrict__ x, float* __restrict__ pm,
                    float* __restrict__ ps, int D, int S) {
  const int row = blockIdx.x / S;
  const int split = blockIdx.x % S;
  const int n = D / S;                      // D divisible by 64 for all shapes
  const float* p = x + (size_t)row * D + (size_t)split * n;
  float m = -INFINITY, s = 0.0f;
  for (int i = threadIdx.x; i < n; i += blockDim.x) {
    float v = p[i];
    if (v > m) { s *= __expf(m - v); m = v; }
    s += __expf(v - m);
  }
  // wave32 butterfly combine of (m, s)
  #pragma unroll
  for (int mask = 16; mask > 0; mask >>= 1) {
    float mo = __shfl_xor(m, mask, 32);
    float so = __shfl_xor(s, mask, 32);
    float M = fmaxf(m, mo);
    s = s * __expf(m - M) + so * __expf(mo - M);
    m = M;
  }
  __shared__ float wm[8], wsum[8];
  const int lane = threadIdx.x & 31, wave = threadIdx.x >> 5;
  if (lane == 0) { wm[wave] = m; wsum[wave] = s; }
  __syncthreads();
  if (threadIdx.x == 0) {
    float M = wm[0], SS = wsum[0];
    #pragma unroll
    for (int w = 1; w < 8; ++w) {
      float MM = fmaxf(M, wm[w]);
      SS = SS * __expf(M - MM) + wsum[w] * __expf(wm[w] - MM);
      M = MM;
    }
    pm[blockIdx.x] = M;
    ps[blockIdx.x] = SS;
  }
}

// ---------------- softmax stats stage 2: combine 64 partials -> m + log s ----------------
__global__ void sm2(const float* __restrict__ pm, const float* __restrict__ ps,
                    float* __restrict__ rowc) {
  const int row = blockIdx.x;
  const int t = threadIdx.x;                // 32 threads
  const float* m_ = pm + row * 64;
  const float* s_ = ps + row * 64;
  float m = m_[t], s = s_[t];
  {
    float mo = m_[t + 32], so = s_[t + 32];
    float M = fmaxf(m, mo);
    s = s * expf(m - M) + so * expf(mo - M);
    m = M;
  }
  #pragma unroll
  for (int mask = 16; mask > 0; mask >>= 1) {
    float mo = __shfl_xor(m, mask, 32);
    float so = __shfl_xor(s, mask, 32);
    float M = fmaxf(m, mo);
    s = s * expf(m - M) + so * expf(mo - M);
    m = M;
  }
  if (t == 0) rowc[row] = m + logf(s);
}

// ---------------- Gram = a @ a^T via V_WMMA_F32_16X16X4_F32 ----------------
// a[i][k] = exp(x[i][k] - rowc[i]); A(16x4) and B(4x16) register layouts
// coincide for A x A^T, so the same registers feed both operands.
// Unrolled 4x: four global_load_b64 in flight, then four WMMAs.
__global__ void gram_kernel(const float* __restrict__ x,
                            const float* __restrict__ rowc,
                            float* __restrict__ G, int D) {
  const int t = threadIdx.x;
  const int lane = t & 31;
  const int wave = __builtin_amdgcn_readfirstlane(t >> 5);   // wave-uniform scalar
  const int row = lane & 15;
  const float c = rowc[row];
  const int koff = (lane < 16) ? 0 : 2;   // A layout: lanes 0-15 K=0,1; 16-31 K=2,3
  const float* p = x + (size_t)row * D + koff;
  const int wavesTotal = gridDim.x * 8;
  const int gw = blockIdx.x * 8 + wave;
  const int kstep = wavesTotal * 16;      // 16 K-values per wave-iteration
  v8f acc = {};
  for (int k = gw * 16; k < D; k += kstep) {   // D % 16 == 0 for all shapes
    float2 x0 = *reinterpret_cast<const float2*>(p + k);
    float2 x1 = *reinterpret_cast<const float2*>(p + k + 4);
    float2 x2 = *reinterpret_cast<const float2*>(p + k + 8);
    float2 x3 = *reinterpret_cast<const float2*>(p + k + 12);
    v2f a0, a1, a2, a3;
    a0.x = __expf(x0.x - c); a0.y = __expf(x0.y - c);
    a1.x = __expf(x1.x - c); a1.y = __expf(x1.y - c);
    a2.x = __expf(x2.x - c); a2.y = __expf(x2.y - c);
    a3.x = __expf(x3.x - c); a3.y = __expf(x3.y - c);
    acc = __builtin_amdgcn_wmma_f32_16x16x4_f32(false, a0, false, a0, (short)0, acc, false, false);
    acc = __builtin_amdgcn_wmma_f32_16x16x4_f32(false, a1, false, a1, (short)0, acc, false, false);
    acc = __builtin_amdgcn_wmma_f32_16x16x4_f32(false, a2, false, a2, (short)0, acc, false, false);
    acc = __builtin_amdgcn_wmma_f32_16x16x4_f32(false, a3, false, a3, (short)0, acc, false, false);
  }
  // C/D layout: VGPR r, lanes 0-15 -> M=r, N=lane; lanes 16-31 -> M=r+8, N=lane-16
  __shared__ float gred[8 * 256];
  const int mbase = (lane < 16) ? 0 : 8;
  #pragma unroll
  for (int r = 0; r < 8; ++r)
    gred[wave * 256 + (r + mbase) * 16 + row] = acc[r];
  __syncthreads();
  float s = 0.0f;
  #pragma unroll
  for (int w = 0; w < 8; ++w) s += gred[w * 256 + t];
  atomicAdd(&G[t], s);
}

// ---------------- reconstruct: mean(1 - cos over channel dim) ----------------
__global__ void rec_kernel(const float* __restrict__ pn, const float* __restrict__ po,
                           float* __restrict__ acc, int C, int HW) {
  const int tid = blockIdx.x * blockDim.x + threadIdx.x;   // grid covers 16*HW exactly
  const int b = tid / HW, pos = tid % HW;
  const float* xp = pn + (size_t)b * C * HW + pos;
  const float* yp = po + (size_t)b * C * HW + pos;
  float d = 0.f, nx = 0.f, ny = 0.f;
  for (int ch = 0; ch < C; ++ch) {
    float xv = xp[(size_t)ch * HW];
    float yv = yp[(size_t)ch * HW];
    d += xv * yv; nx += xv * xv; ny += yv * yv;
  }
  float cosv = d / (fmaxf(sqrtf(nx), 1e-8f) * fmaxf(sqrtf(ny), 1e-8f));
  float local = 1.0f - cosv;
  __shared__ float red[256];
  red[threadIdx.x] = local; __syncthreads();
  for (int s = 128; s > 0; s >>= 1) {
    if (threadIdx.x < s) red[threadIdx.x] += red[threadIdx.x + s];
    __syncthreads();
  }
  if (threadIdx.x == 0) atomicAdd(acc, red[0]);
}

// ---------------- contrast: per-row dot/|x|^2/|y|^2 partials ----------------
__global__ void con_kernel(const float* __restrict__ xf, const float* __restrict__ yf,
                           float* __restrict__ conOut, int D, int S) {
  const int row = blockIdx.x / S, split = blockIdx.x % S;
  const int n = D / S;
  const float* x = xf + (size_t)row * D + (size_t)split * n;
  const float* y = yf + (size_t)row * D + (size_t)split * n;
  float d = 0.f, a2 = 0.f, b2 = 0.f;
  for (int i = threadIdx.x; i < n; i += blockDim.x) {
    float xv = x[i], yv = y[i];
    d += xv * yv; a2 += xv * xv; b2 += yv * yv;
  }
  __shared__ float red[256];
  float vals[3] = {d, a2, b2};
  #pragma unroll
  for (int c = 0; c < 3; ++c) {
    red[threadIdx.x] = vals[c]; __syncthreads();
    for (int s = 128; s > 0; s >>= 1) {
      if (threadIdx.x < s) red[threadIdx.x] += red[threadIdx.x + s];
      __syncthreads();
    }
    if (threadIdx.x == 0) atomicAdd(&conOut[row * 3 + c], red[0]);
    __syncthreads();
  }
}

// ---------------- Sinkhorn on 16x16 + final combine ----------------
__device__ __forceinline__ float softmin_row(float Cv, float hj, float eps) {
  // per-thread (i,j); logsumexp over j within 16-lane group
  float term = hj - Cv / eps;
  float m = term;
  #pragma unroll
  for (int mask = 8; mask > 0; mask >>= 1)
    m = fmaxf(m, __shfl_xor(m, mask, 16));
  float e = expf(term - m);
  #pragma unroll
  for (int mask = 8; mask > 0; mask >>= 1)
    e += __shfl_xor(e, mask, 16);
  return -eps * (logf(e) + m);
}

__global__ void final_kernel(const float* __restrict__ G,
                             const float* __restrict__ rec,
                             const float* __restrict__ con,
                             const int* __restrict__ perm,
                             float* __restrict__ out) {
  const int t = threadIdx.x;
  const int i = t >> 4, j = t & 15;
  __shared__ float f_ab[16], g_ba[16], f_aa[16], g_bb[16];
  __shared__ float rowsum[16];
  __shared__ int sperm[16];
  __shared__ float ssot_acc;
  if (t < 16) sperm[t] = perm[t];
  if (t == 0) ssot_acc = 0.0f;
  __syncthreads();
  const float wlog = -logf(16.0f);
  const float start = 2.0f * logf(1e7f);
  const float step  = 2.0f * logf(0.95f);
  const float stop  = 2.0f * logf(0.05f);
  const int nsteps  = (int)ceilf((stop - start) / step);   // ~373 arange entries

  for (int k = 0; k < 3; ++k) {
    const float* Gk = G + k * 256;
    const int pi = sperm[i], pj = sperm[j];
    const float dI  = Gk[i * 16 + i],   dJ  = Gk[j * 16 + j];
    const float dPi = Gk[pi * 16 + pi], dPj = Gk[pj * 16 + pj];
    const float Cxy = 0.5f * fmaxf(dI + dPj - 2.f * Gk[i * 16 + pj], 0.f);
    const float Cyx = 0.5f * fmaxf(dPi + dJ - 2.f * Gk[pi * 16 + j], 0.f);
    const float Cxx = 0.5f * fmaxf(dI + dJ - 2.f * Gk[i * 16 + j], 0.f);
    const float Cyy = 0.5f * fmaxf(dPi + dPj - 2.f * Gk[pi * 16 + pj], 0.f);

    // init with eps0 = diameter^p
    {
      const float eps0 = 1e14f;
      float v0 = softmin_row(Cxy, wlog, eps0);
      float v1 = softmin_row(Cyx, wlog, eps0);
      float v2 = softmin_row(Cxx, wlog, eps0);
      float v3 = softmin_row(Cyy, wlog, eps0);
      __syncthreads();
      if (j == 0) { f_ab[i] = v0; g_ba[i] = v1; f_aa[i] = v2; g_bb[i] = v3; }
      __syncthreads();
    }
    // scan over [1e14, arange(...), blur^p] with symmetric averaged updates
    for (int it = -1; it <= nsteps; ++it) {
      const float eps = (it < 0) ? 1e14f
                      : ((it < nsteps) ? expf(start + step * (float)it) : 0.0025f);
      float ft  = softmin_row(Cxy, wlog + g_ba[j] / eps, eps);
      float gt  = softmin_row(Cyx, wlog + f_ab[j] / eps, eps);
      float fat = softmin_row(Cxx, wlog + f_aa[j] / eps, eps);
      float gbt = softmin_row(Cyy, wlog + g_bb[j] / eps, eps);
      __syncthreads();
      if (j == 0) {
        f_ab[i] = 0.5f * (f_ab[i] + ft);
        g_ba[i] = 0.5f * (g_ba[i] + gt);
        f_aa[i] = 0.5f * (f_aa[i] + fat);
        g_bb[i] = 0.5f * (g_bb[i] + gbt);
      }
      __syncthreads();
    }
    // last extrapolation (no averaging)
    {
      const float eps = 0.0025f;
      float fxy = softmin_row(Cxy, wlog + g_ba[j] / eps, eps);
      float gyx = softmin_row(Cyx, wlog + f_ab[j] / eps, eps);
      float fxx = softmin_row(Cxx, wlog + f_aa[j] / eps, eps);
      float gyy = softmin_row(Cyy, wlog + g_bb[j] / eps, eps);
      __syncthreads();
      if (j == 0) rowsum[i] = (fxy - fxx) + (gyx - gyy);
      __syncthreads();
      if (t == 0) {
        float s = 0.f;
        #pragma unroll
        for (int r = 0; r < 16; ++r) s += rowsum[r];
        ssot_acc += s / 16.0f;
      }
      __syncthreads();
    }
  }

  if (t == 0) {
    float lrec = rec[0] / 65536.f + rec[1] / 16384.f + rec[2] / 4096.f;
    float lcon = 0.f;
    for (int k = 0; k < 3; ++k) {
      for (int r = 0; r < 16; ++r) {
        float d  = con[k * 48 + r * 3 + 0];
        float a2 = con[k * 48 + r * 3 + 1];
        float b2 = con[k * 48 + r * 3 + 2];
        float c = d / (fmaxf(sqrtf(a2), 1e-8f) * fmaxf(sqrtf(b2), 1e-8f));
        lcon += fmaxf(c - 0.5f, 0.f);
      }
    }
    lcon /= 16.f;
    out[0] = (ssot_acc + 0.01f * lrec + 0.1f * lcon) / 1.11f;
  }
}

extern "C" void kernel_launch(void* const* d_in, const int* in_sizes, int n_in,
                              void* d_out, int out_size, void* d_ws, size_t ws_size,
                              hipStream_t stream) {
  (void)in_sizes; (void)n_in; (void)out_size; (void)ws_size;
  const float* nf[3]  = {(const float*)d_in[0], (const float*)d_in[1], (const float*)d_in[2]};
  const float* pnf[3] = {(const float*)d_in[3], (const float*)d_in[4], (const float*)d_in[5]};
  const float* pof[3] = {(const float*)d_in[6], (const float*)d_in[7], (const float*)d_in[8]};
  const int* perm = (const int*)d_in[9];
  float* ws = (float*)d_ws;
  float* out = (float*)d_out;

  const int Cs[3] = {256, 512, 1024};
  const int Hs[3] = {64, 32, 16};

  init_ws<<<1, 256, 0, stream>>>(ws);

  for (int k = 0; k < 3; ++k) {
    const int HW = Hs[k] * Hs[k];
    const int D = Cs[k] * HW;                 // per-row flattened length
    // softmax row stats (ssot)
    sm1<<<16 * 64, 256, 0, stream>>>(pof[k], ws + WS_PM + k * 1024,
                                     ws + WS_PS + k * 1024, D, 64);
    sm2<<<16, 32, 0, stream>>>(ws + WS_PM + k * 1024, ws + WS_PS + k * 1024,
                               ws + WS_ROWC + k * 16);
    // Gram of softmax rows via f32 WMMA (unrolled 4x)
    gram_kernel<<<256, 256, 0, stream>>>(pof[k], ws + WS_ROWC + k * 16,
                                         ws + WS_G + k * 256, D);
    // reconstruct loss partials
    rec_kernel<<<(16 * HW) / 256, 256, 0, stream>>>(pnf[k], pof[k],
                                                    ws + WS_REC + k, Cs[k], HW);
    // contrast loss partials
    con_kernel<<<16 * 64, 256, 0, stream>>>(nf[k], pof[k],
                                            ws + WS_CON + k * 48, D, 64);
  }

  final_kernel<<<1, 256, 0, stream>>>(ws + WS_G, ws + WS_REC, ws + WS_CON, perm, out);
}